// RGCNConcatModel_39505109188801
// MI455X (gfx1250) — compile-verified
//
#include <hip/hip_runtime.h>
#include <hip/hip_bf16.h>

typedef __attribute__((ext_vector_type(2))) float v2f;
typedef __attribute__((ext_vector_type(8))) float v8f;
typedef __attribute__((ext_vector_type(4))) int   v4i;

#define NB      32
#define LLEN    1024
#define LMD     1024
#define DD      128
#define RR      3
#define NLAYERS 2
#define CC      8
#define NN      (NB * LLEN)   // 32768 nodes
#define EE      524288        // edges

// ---------------------------------------------------------------------------
// gfx1250 async global->LDS staging (ASYNCcnt path), with safe fallback.
// Builtin signature (from hipcc diagnostic): param0 = v4i addrspace(1)*,
// param1 = v4i addrspace(3)*, then imm offset, imm cpol.
// ---------------------------------------------------------------------------
#if defined(__has_builtin)
#if __has_builtin(__builtin_amdgcn_global_load_async_to_lds_b128) && \
    __has_builtin(__builtin_amdgcn_s_wait_asynccnt)
#define HAVE_ASYNC_LDS 1
#endif
#endif

#define GLOBAL_AS __attribute__((address_space(1)))
#define LDS_AS    __attribute__((address_space(3)))

__device__ __forceinline__ void stage16(const float* g, float* l)
{
#ifdef HAVE_ASYNC_LDS
    float* gn = const_cast<float*>(g);
    __builtin_amdgcn_global_load_async_to_lds_b128(
        (GLOBAL_AS v4i*)gn, (LDS_AS v4i*)l, 0, 0);
#else
    *(float4*)l = *(const float4*)g;
#endif
}

__device__ __forceinline__ void stage_wait()
{
#ifdef HAVE_ASYNC_LDS
    __builtin_amdgcn_s_wait_asynccnt(0);
#endif
}

// ---------------------------------------------------------------------------
// LM head: lm = LayerNorm(ReLU(output @ lm_W^T + lm_b))
// Block: 16 rows x 128 cols, 8 waves, each wave one 16x16 WMMA tile.
// A (16x64) and B (128x64, shared by all waves) staged in LDS per k-chunk;
// the WMMA inner loop touches only LDS.
// ---------------------------------------------------------------------------
__global__ __launch_bounds__(256) void k_lmhead(const float* __restrict__ out_in,
                                                const float* __restrict__ lmW,
                                                const float* __restrict__ lmb,
                                                const float* __restrict__ lng,
                                                const float* __restrict__ lnb,
                                                float* __restrict__ lm_out)
{
    __shared__ float At[16][68];     // 16x64 A tile, padded (conflict-free)
    __shared__ float Bt[128][68];    // 128x64 lm_W slab (K-contiguous rows)
    __shared__ float Lt[16][132];    // 16x128 result tile, padded

    const int tid  = threadIdx.x;
    const int lane = tid & 31;
    const int wv   = tid >> 5;       // 0..7
    const int g    = lane >> 4;      // half-wave: k-pair selector
    const int l16  = lane & 15;
    const int m0   = blockIdx.x * 16;
    const int n0   = wv * 16;

    v8f acc = {};

    for (int k0 = 0; k0 < LMD; k0 += 64) {
        // ---- stage A tile: 16 rows x 64 K (1 x b128 per thread) ----
        {
            const int row = tid >> 4, quad = tid & 15;
            stage16(out_in + (size_t)(m0 + row) * LMD + k0 + quad * 4,
                    &At[row][quad * 4]);
        }
        // ---- stage B slab: lm_W[0:128][k0:k0+64] (8 x b128 per thread) ----
        #pragma unroll
        for (int j = 0; j < 8; ++j) {
            const int idx = tid + 256 * j;          // 0..2047
            const int h = idx >> 4, quad = idx & 15;
            stage16(lmW + (size_t)h * LMD + k0 + quad * 4, &Bt[h][quad * 4]);
        }
        stage_wait();
        __syncthreads();

        #pragma unroll
        for (int kk = 0; kk < 64; kk += 4) {
            v2f a, b;
            const float2 aw = *(const float2*)&At[l16][kk + 2 * g];
            const float2 bw = *(const float2*)&Bt[n0 + l16][kk + 2 * g];
            a.x = aw.x; a.y = aw.y;
            b.x = bw.x; b.y = bw.y;
            acc = __builtin_amdgcn_wmma_f32_16x16x4_f32(false, a, false, b,
                                                        (short)0, acc, false, false);
        }
        __syncthreads();
    }

    // bias + ReLU -> LDS tile (C layout: VGPR i -> M = i + 8*g, N = l16)
    const float bias = lmb[n0 + l16];
    #pragma unroll
    for (int i = 0; i < 8; ++i)
        Lt[i + 8 * g][n0 + l16] = fmaxf(acc[i] + bias, 0.0f);
    __syncthreads();

    // LayerNorm over D=128: 16 threads per row, shuffle-reduce width 16 (wave32-safe)
    const int row = tid >> 4, sub = tid & 15;
    float s = 0.0f, sq = 0.0f;
    #pragma unroll
    for (int j = 0; j < 8; ++j) {
        const float v = Lt[row][sub + 16 * j];
        s += v; sq += v * v;
    }
    #pragma unroll
    for (int off = 8; off >= 1; off >>= 1) {
        s  += __shfl_xor(s,  off, 16);
        sq += __shfl_xor(sq, off, 16);
    }
    const float mu   = s * (1.0f / 128.0f);
    const float var  = sq * (1.0f / 128.0f) - mu * mu;
    const float rstd = rsqrtf(var + 1e-5f);
    #pragma unroll
    for (int j = 0; j < 8; ++j) {
        const int col = sub + 16 * j;
        lm_out[(size_t)(m0 + row) * DD + col] =
            (Lt[row][col] - mu) * rstd * lng[col] + lnb[col];
    }
}

// ---------------------------------------------------------------------------
// Per-(node,relation) edge counts
// ---------------------------------------------------------------------------
__global__ __launch_bounds__(256) void k_count(const int* __restrict__ ei,
                                               const int* __restrict__ et,
                                               unsigned* __restrict__ cnt)
{
    const int e = blockIdx.x * 256 + threadIdx.x;
    if (e < EE) {
        const unsigned seg = (unsigned)ei[EE + e] * RR + (unsigned)et[e];
        atomicAdd(&cnt[seg], 1u);
    }
}

// ---------------------------------------------------------------------------
// Edge scatter: sums[(dst*R+rel)*D + d] += x[src*D + d]   (2 edges / block)
// ---------------------------------------------------------------------------
__global__ __launch_bounds__(256) void k_scatter(const float* __restrict__ x,
                                                 const int* __restrict__ ei,
                                                 const int* __restrict__ et,
                                                 float* __restrict__ sums)
{
    const int e = blockIdx.x * 2 + (threadIdx.x >> 7);
    const int d = threadIdx.x & 127;
    const int s = ei[e];
    const unsigned seg = (unsigned)ei[EE + e] * RR + (unsigned)et[e];
    atomicAdd(&sums[(size_t)seg * DD + d], x[(size_t)s * DD + d]);
}

// ---------------------------------------------------------------------------
// RGCN transform: x_out = ReLU([mean_r0|mean_r1|mean_r2|x] @ [Wrel;Wroot] + b)
// Virtual K = 512; mean scaling folded into the A staging. B slab staged in
// LDS with 144-float row pitch (2*144 mod 64 == 32 -> the two half-wave row
// reads hit disjoint bank sets).
// ---------------------------------------------------------------------------
__global__ __launch_bounds__(256) void k_transform(const float* __restrict__ sums,
                                                   const unsigned* __restrict__ cnt,
                                                   const float* __restrict__ xin,
                                                   const float* __restrict__ Wrel,   // [384,128]
                                                   const float* __restrict__ Wroot,  // [128,128]
                                                   const float* __restrict__ convb,  // [128]
                                                   float* __restrict__ xout)
{
    __shared__ float At[16][68];
    __shared__ float Bt[64][144];   // rows k0..k0+63 of [Wrel;Wroot], all 128 cols

    const int tid  = threadIdx.x;
    const int lane = tid & 31;
    const int wv   = tid >> 5;
    const int g    = lane >> 4;
    const int l16  = lane & 15;
    const int m0   = blockIdx.x * 16;
    const int n0   = wv * 16;

    v8f acc = {};

    for (int k0 = 0; k0 < 512; k0 += 64) {
        // ---- stage virtual A (scaled relation means | root x) ----
        {
            const int row = tid >> 4, quad = tid & 15;
            const int k = k0 + quad * 4;
            float4 v;
            if (k < 384) {
                v = *(const float4*)(sums + (size_t)(m0 + row) * 384 + k);
                const unsigned c = cnt[(m0 + row) * RR + (k >> 7)];
                const float inv = 1.0f / fmaxf((float)c, 1.0f);
                v.x *= inv; v.y *= inv; v.z *= inv; v.w *= inv;
            } else {
                v = *(const float4*)(xin + (size_t)(m0 + row) * DD + (k - 384));
            }
            *(float4*)&At[row][quad * 4] = v;
        }
        // ---- stage B slab: rows k0..k0+63, all 128 cols (8 x b128/thread) ----
        #pragma unroll
        for (int j = 0; j < 8; ++j) {
            const int idx = tid + 256 * j;           // 0..2047
            const int row = idx >> 5, qc = (idx & 31) * 4;
            const int kg = k0 + row;
            const float* src = (kg < 384) ? (Wrel + (size_t)kg * DD + qc)
                                          : (Wroot + (size_t)(kg - 384) * DD + qc);
            stage16(src, &Bt[row][qc]);
        }
        stage_wait();
        __syncthreads();

        #pragma unroll
        for (int kk = 0; kk < 64; kk += 4) {
            v2f a, b;
            const float2 aw = *(const float2*)&At[l16][kk + 2 * g];
            a.x = aw.x; a.y = aw.y;
            b.x = Bt[kk + 2 * g][n0 + l16];
            b.y = Bt[kk + 2 * g + 1][n0 + l16];
            acc = __builtin_amdgcn_wmma_f32_16x16x4_f32(false, a, false, b,
                                                        (short)0, acc, false, false);
        }
        __syncthreads();
    }

    const int col = n0 + l16;
    const float bias = convb[col];
    #pragma unroll
    for (int i = 0; i < 8; ++i) {
        const int m = i + 8 * g;
        xout[(size_t)(m0 + m) * DD + col] = fmaxf(acc[i] + bias, 0.0f);
    }
}

// ---------------------------------------------------------------------------
// Classifier + fused log-softmax + masked NLL accumulation
// ---------------------------------------------------------------------------
__global__ __launch_bounds__(256) void k_classifier(const float* __restrict__ lm,
                                                    const float* __restrict__ gx,
                                                    const float* __restrict__ clsW,
                                                    const float* __restrict__ clsb,
                                                    const int* __restrict__ amask,
                                                    const int* __restrict__ labels,
                                                    float* __restrict__ logits,
                                                    float* __restrict__ accum)
{
    __shared__ float Ws[CC * 2 * DD];  // 8 x 256
    __shared__ float bs[CC];
    const int tid = threadIdx.x;
    for (int i = tid; i < CC * 2 * DD; i += 256) Ws[i] = clsW[i];
    if (tid < CC) bs[tid] = clsb[tid];
    __syncthreads();

    const int n = blockIdx.x * 256 + tid;
    float a[CC];
    #pragma unroll
    for (int c = 0; c < CC; ++c) a[c] = bs[c];

    const float* lrow = lm + (size_t)n * DD;
    const float* grow = gx + (size_t)n * DD;
    for (int j = 0; j < DD; ++j) {
        const float lv = lrow[j], gv = grow[j];
        #pragma unroll
        for (int c = 0; c < CC; ++c)
            a[c] += lv * Ws[c * 256 + j] + gv * Ws[c * 256 + DD + j];
    }

    float mx = a[0];
    #pragma unroll
    for (int c = 1; c < CC; ++c) mx = fmaxf(mx, a[c]);
    float se = 0.0f;
    #pragma unroll
    for (int c = 0; c < CC; ++c) se += expf(a[c] - mx);
    const float lse = mx + logf(se);

    #pragma unroll
    for (int c = 0; c < CC; ++c) logits[(size_t)n * CC + c] = a[c];

    const int lab = labels[n];
    float alab = 0.0f;
    #pragma unroll
    for (int c = 0; c < CC; ++c) alab = (c == lab) ? a[c] : alab;

    const bool valid = (amask[n] == 1);
    float vn = valid ? (lse - alab) : 0.0f;
    float vc = valid ? 1.0f : 0.0f;
    #pragma unroll
    for (int off = 16; off >= 1; off >>= 1) {
        vn += __shfl_xor(vn, off, 32);
        vc += __shfl_xor(vc, off, 32);
    }
    if ((tid & 31) == 0) {
        atomicAdd(&accum[0], vn);
        atomicAdd(&accum[1], vc);
    }
}

__global__ void k_finalize(const float* __restrict__ accum, float* __restrict__ loss)
{
    loss[0] = accum[0] / fmaxf(accum[1], 1.0f);
}

// ---------------------------------------------------------------------------
extern "C" void kernel_launch(void* const* d_in, const int* in_sizes, int n_in,
                              void* d_out, int out_size, void* d_ws, size_t ws_size,
                              hipStream_t stream)
{
    (void)in_sizes; (void)n_in; (void)out_size; (void)ws_size;

    const float* output = (const float*)d_in[0];
    const int*   ei     = (const int*)d_in[1];
    const int*   et     = (const int*)d_in[2];
    const int*   amask  = (const int*)d_in[3];
    const int*   labels = (const int*)d_in[4];
    const float* lmW    = (const float*)d_in[5];
    const float* lmb    = (const float*)d_in[6];
    const float* lng    = (const float*)d_in[7];
    const float* lnb    = (const float*)d_in[8];
    const float* Wrel   = (const float*)d_in[9];
    const float* Wroot  = (const float*)d_in[10];
    const float* convb  = (const float*)d_in[11];
    const float* clsW   = (const float*)d_in[12];
    const float* clsb   = (const float*)d_in[13];

    // workspace layout (bytes)
    char* ws = (char*)d_ws;
    const size_t XB = (size_t)NN * DD * 4;            // 16 MB per node-feature buffer
    const size_t SB = (size_t)NN * RR * DD * 4;       // 48 MB relation sums
    float*    lm    = (float*)(ws);
    float*    x0    = (float*)(ws + XB);
    float*    x1    = (float*)(ws + 2 * XB);
    float*    sums  = (float*)(ws + 3 * XB);
    unsigned* cnt   = (unsigned*)(ws + 3 * XB + SB);
    float*    accum = (float*)(ws + 3 * XB + SB + (size_t)NN * RR * 4);

    float* loss_out   = (float*)d_out;
    float* logits_out = (float*)d_out + 1;

    // zero counts + loss accumulators (one memset)
    (void)hipMemsetAsync(cnt, 0, (size_t)NN * RR * 4 + 8, stream);

    k_count<<<EE / 256, 256, 0, stream>>>(ei, et, cnt);
    k_lmhead<<<NN / 16, 256, 0, stream>>>(output, lmW, lmb, lng, lnb, lm);

    const float* xin = lm;
    float* xbuf[2] = { x0, x1 };
    for (int layer = 0; layer < NLAYERS; ++layer) {
        (void)hipMemsetAsync(sums, 0, SB, stream);
        k_scatter<<<EE / 2, 256, 0, stream>>>(xin, ei, et, sums);
        k_transform<<<NN / 16, 256, 0, stream>>>(
            sums, cnt, xin,
            Wrel + (size_t)layer * RR * DD * DD,
            Wroot + (size_t)layer * DD * DD,
            convb + (size_t)layer * DD,
            xbuf[layer]);
        xin = xbuf[layer];
    }

    k_classifier<<<NN / 256, 256, 0, stream>>>(lm, x1, clsW, clsb, amask, labels,
                                               logits_out, accum);
    k_finalize<<<1, 1, 0, stream>>>(accum, loss_out);
}